// ParallelConvSSM_13091060318284
// MI455X (gfx1250) — compile-verified
//
#include <hip/hip_runtime.h>
#include <math.h>

// ---------------------------------------------------------------------------
// ParallelConvSSM on gfx1250:  out = ifft2( fft2(x) * G ),
//   G[u,v,c] = DFT2(pad(B_kernel)) * sum_{k=0..7} DFT2(pad(0.9*tanh(A)))^k / (H*W)
// All DFT/IDFTs are f32 WMMA GEMMs (V_WMMA_F32_16X16X4_F32), M=K=128.
// B panels are staged in LDS (transposed, stride 132 -> conflict-free b64
// reads); DFT sign is folded into staging so the hot loop is loads + WMMA only.
// ---------------------------------------------------------------------------

typedef __attribute__((ext_vector_type(2))) float v2f;
typedef __attribute__((ext_vector_type(8))) float v8f;

#define HDIM   128
#define WDIM   128
#define CDIM   256
#define NBATCH 2
#define KS     7
#define TAPS   49
#define PLANE  (HDIM * WDIM * CDIM)   // 4,194,304 floats per batch
#define ROWLEN (WDIM * CDIM)          // 32,768
#define PI_64  0.04908738521234052f   // pi/64
#define CSTRIDE 132                   // LDS per-column stride (floats)

__device__ __forceinline__ v8f wmma_f32(v2f a, v2f b, v8f c) {
  // 8 args: (neg_a, A, neg_b, B, c_mod, C, reuse_a, reuse_b); f32 has no A/B neg
  return __builtin_amdgcn_wmma_f32_16x16x4_f32(false, a, false, b, (short)0, c,
                                               false, false);
}

// ---------------------------------------------------------------------------
// A_stable = 0.9 * tanh(A_kernel)
__global__ void prep_tanh(const float* __restrict__ Ak, float* __restrict__ As,
                          int n) {
  int i = blockIdx.x * blockDim.x + threadIdx.x;
  if (i < n) As[i] = 0.9f * tanhf(Ak[i]);
}

// DFT twiddle matrices: Fc[u*128+h] = cos(2*pi*u*h/128), Fs = sin(...)
__global__ void build_dft(float* __restrict__ Fc, float* __restrict__ Fs) {
  int idx = blockIdx.x * blockDim.x + threadIdx.x;  // 16384 threads
  int u = idx >> 7, h = idx & 127;
  int t = (u * h) & 127;
  float th = (float)t * PI_64;
  Fc[idx] = cosf(th);
  Fs[idx] = sinf(th);
}

// G[u][v][c] = B_f * (sum_{k=0..7} A_f^k) / 16384, with the reference's
// center-wrapped padding: phase = exp(-2*pi*i*(u*(i-3) + v*(j-3))/128).
__global__ void build_filter(const float* __restrict__ As,
                             const float* __restrict__ Bk,
                             float* __restrict__ Gre, float* __restrict__ Gim) {
  unsigned t = blockIdx.x * blockDim.x + threadIdx.x;  // 128*128*256 threads
  int c = t & (CDIM - 1);
  int v = (t >> 8) & 127;
  int u = t >> 15;

  float wur[KS], wui[KS], wvr[KS], wvi[KS];
#pragma unroll
  for (int i = 0; i < KS; ++i) {
    int e = u * (i - 3);
    int m = ((e % 128) + 128) & 127;
    float s, co;
    __sincosf((float)m * PI_64, &s, &co);
    wur[i] = co; wui[i] = -s;
    e = v * (i - 3);
    m = ((e % 128) + 128) & 127;
    __sincosf((float)m * PI_64, &s, &co);
    wvr[i] = co; wvi[i] = -s;
  }

  float ar = 0.f, ai = 0.f, br = 0.f, bi = 0.f;
  const float* ap = As + c * TAPS;
  const float* bp = Bk + c * TAPS;
#pragma unroll
  for (int i = 0; i < KS; ++i) {
    float xr = wur[i], xi = wui[i];
#pragma unroll
    for (int j = 0; j < KS; ++j) {
      float er = xr * wvr[j] - xi * wvi[j];
      float ei = xr * wvi[j] + xi * wvr[j];
      float a = ap[i * KS + j];
      ar += a * er; ai += a * ei;
      float b = bp[i * KS + j];
      br += b * er; bi += b * ei;
    }
  }
  // S = 1 + A*(1 + A*(...)) = sum_{k=0..7} A^k   (7 complex Horner steps)
  float sr = 1.f, si = 0.f;
#pragma unroll
  for (int k = 0; k < 7; ++k) {
    float nr = ar * sr - ai * si + 1.f;
    float ni = ar * si + ai * sr;
    sr = nr; si = ni;
  }
  const float inv = 1.0f / 16384.0f;  // ifft2 normalization folded in
  size_t g = ((size_t)u * WDIM + v) * CDIM + c;
  Gre[g] = (br * sr - bi * si) * inv;
  Gim[g] = (br * si + bi * sr) * inv;
}

// Pointwise complex multiply in frequency domain (in place on Y).
__global__ void apply_filter(float* __restrict__ Yre, float* __restrict__ Yim,
                             const float* __restrict__ Gre,
                             const float* __restrict__ Gim) {
  unsigned t = blockIdx.x * blockDim.x + threadIdx.x;  // exactly PLANE threads
  float yr = Yre[t], yi = Yim[t], gr = Gre[t], gi = Gim[t];
  Yre[t] = yr * gr - yi * gi;
  Yim[t] = yr * gi + yi * gr;
}

// ---------------------------------------------------------------------------
// Complex DFT GEMM:  Out = (Fc + i*sign*Fs) @ (Xre + i*Xim),  M=K=128.
//   out_re = Fc@Xre - sign*Fs@Xim
//   out_im = sign*(Fs@Xre + Fc@(sign*Xim))        (sign^2 = 1)
// The staged imaginary panel already carries `sign`, so the K-loop is pure
// loads + WMMA; signs are resolved in the epilogue from 4 accumulators.
//
// LDS panel layout: transposed [col][row], stride CSTRIDE=132 floats.
//   - lane's B operand {B[k+koff][col], B[k+koff+1][col]} = one aligned
//     ds_load_b64 at col*132*4 + (k+koff)*4 (immediate offsets only)
//   - stride 132: lane c -> banks {4c+r, 4c+r+1}, upper half-wave (row +2) ->
//     {4c+r+2, 4c+r+3}: all 64 banks exactly once, conflict-free.
// WMMA f32 16x16x4 operand layout (ISA 7.12.2):
//   A: lane L<16 -> row m0+L, K={k,k+1}; lane>=16 -> same rows, K={k+2,k+3}
//   B: VGPR0 = row k+koff, VGPR1 = row k+koff+1, koff = 2*(lane>=16)
//   C/D: VGPR r -> row m0 + r + 8*(lane>=16), col n0 + (lane&15)
template <bool HAS_XIM, bool HAS_OIM>
__global__ void __launch_bounds__(256)
dft_cgemm(const float* __restrict__ Fc, const float* __restrict__ Fs,
          const float* __restrict__ Xre, const float* __restrict__ Xim,
          float* __restrict__ Ore, float* __restrict__ Oim,
          int ld, long sliceInStride, long sliceOutStride, float sign) {
  __shared__ float sRe[16 * CSTRIDE];
  __shared__ float sIm[HAS_XIM ? 16 * CSTRIDE : 1];

  const int lane = threadIdx.x & 31;
  const int wave = threadIdx.x >> 5;
  const int m0 = wave << 4;
  const int n0 = blockIdx.x << 4;
  const long sIn = (long)blockIdx.z * sliceInStride;
  const long sOut = (long)blockIdx.z * sliceOutStride;

  // ---- stage B panel (K=128 x N=16) into LDS, transposed, sign folded in ----
  for (int e = threadIdx.x; e < 128 * 16; e += 256) {
    int row = e >> 4;       // k index 0..127
    int colL = e & 15;      // local column
    long gidx = sIn + (long)row * ld + n0 + colL;
    sRe[colL * CSTRIDE + row] = Xre[gidx];
    if (HAS_XIM) sIm[colL * CSTRIDE + row] = sign * Xim[gidx];
  }
  __syncthreads();

  // ---- K loop: pure loads + WMMA ----
  const int koff = (lane >> 4) << 1;               // 0 or 2
  const int mrow = m0 + (lane & 15);
  const float* aCp = Fc + mrow * 128;
  const float* aSp = Fs + mrow * 128;
  const float* bRp = &sRe[(lane & 15) * CSTRIDE];
  const float* bIp = HAS_XIM ? &sIm[(lane & 15) * CSTRIDE] : nullptr;

  v8f acc_cr = {0.f, 0.f, 0.f, 0.f, 0.f, 0.f, 0.f, 0.f};  // Fc @ Xre
  v8f acc_sr = {0.f, 0.f, 0.f, 0.f, 0.f, 0.f, 0.f, 0.f};  // Fs @ Xre
  v8f acc_ci = {0.f, 0.f, 0.f, 0.f, 0.f, 0.f, 0.f, 0.f};  // Fc @ (s*Xim)
  v8f acc_si = {0.f, 0.f, 0.f, 0.f, 0.f, 0.f, 0.f, 0.f};  // Fs @ (s*Xim)

#pragma unroll 8
  for (int k = 0; k < 128; k += 4) {
    v2f ac = *(const v2f*)(aCp + k + koff);
    v2f as = *(const v2f*)(aSp + k + koff);
    v2f br = *(const v2f*)(bRp + k + koff);
    acc_cr = wmma_f32(ac, br, acc_cr);
    if (HAS_OIM) acc_sr = wmma_f32(as, br, acc_sr);
    if (HAS_XIM) {
      v2f bi = *(const v2f*)(bIp + k + koff);
      acc_si = wmma_f32(as, bi, acc_si);
      if (HAS_OIM) acc_ci = wmma_f32(ac, bi, acc_ci);
    }
  }

  // ---- epilogue: resolve signs, store ----
  const int rowAdd = (lane >> 4) << 3;
  const int col = n0 + (lane & 15);
#pragma unroll
  for (int r = 0; r < 8; ++r) {
    long idx = sOut + (long)(m0 + r + rowAdd) * ld + col;
    float vre = acc_cr[r];
    if (HAS_XIM) vre -= acc_si[r];   // acc_si = sign*Fs@Xim
    Ore[idx] = vre;
    if (HAS_OIM) {
      float vim = acc_sr[r];
      if (HAS_XIM) vim += acc_ci[r];
      Oim[idx] = sign * vim;         // sign*(Fs@Xre + Fc@(sign*Xim))
    }
  }
}

// ---------------------------------------------------------------------------
extern "C" void kernel_launch(void* const* d_in, const int* in_sizes, int n_in,
                              void* d_out, int out_size, void* d_ws,
                              size_t ws_size, hipStream_t stream) {
  const float* x = (const float*)d_in[0];
  const float* Ak = (const float*)d_in[1];
  const float* Bk = (const float*)d_in[2];
  float* out = (float*)d_out;

  float* w = (float*)d_ws;
  float* Fc = w;   w += 16384;        // cos twiddles (128x128)
  float* Fs = w;   w += 16384;        // sin twiddles
  float* As = w;   w += 12544;        // 0.9*tanh(A_kernel), C x 49
  float* Gre = w;  w += PLANE;        // effective filter, [u][v][c]
  float* Gim = w;  w += PLANE;
  float* Y1re = w; w += PLANE;        // per-batch intermediates
  float* Y1im = w; w += PLANE;
  float* Y2re = w; w += PLANE;
  float* Y2im = w; w += PLANE;        // total ~101 MB (fits in 192 MB L2)

  prep_tanh<<<(12544 + 255) / 256, 256, 0, stream>>>(Ak, As, 12544);
  build_dft<<<64, 256, 0, stream>>>(Fc, Fs);
  build_filter<<<PLANE / 256, 256, 0, stream>>>(As, Bk, Gre, Gim);

  for (int b = 0; b < NBATCH; ++b) {
    const float* xb = x + (size_t)b * PLANE;
    float* ob = out + (size_t)b * PLANE;

    // S1: forward DFT over h (rows).  Real input -> complex Y1[u][w*C+c].
    dft_cgemm<false, true><<<dim3(ROWLEN / 16, 1, 1), 256, 0, stream>>>(
        Fc, Fs, xb, nullptr, Y1re, Y1im, ROWLEN, 0, 0, -1.0f);
    // S2: forward DFT over w, one slice per u.  Y1[u] is (W x C), ld = C.
    dft_cgemm<true, true><<<dim3(CDIM / 16, 1, HDIM), 256, 0, stream>>>(
        Fc, Fs, Y1re, Y1im, Y2re, Y2im, CDIM, ROWLEN, ROWLEN, -1.0f);
    // S3: multiply by effective SSM filter in frequency domain.
    apply_filter<<<PLANE / 256, 256, 0, stream>>>(Y2re, Y2im, Gre, Gim);
    // S4: inverse DFT over v, one slice per u (1/N folded into G).
    dft_cgemm<true, true><<<dim3(CDIM / 16, 1, HDIM), 256, 0, stream>>>(
        Fc, Fs, Y2re, Y2im, Y1re, Y1im, CDIM, ROWLEN, ROWLEN, +1.0f);
    // S5: inverse DFT over u, real part only -> output.
    dft_cgemm<true, false><<<dim3(ROWLEN / 16, 1, 1), 256, 0, stream>>>(
        Fc, Fs, Y1re, Y1im, ob, nullptr, ROWLEN, 0, 0, +1.0f);
  }
}